// MultiTaskLossWrapper_79980880986291
// MI455X (gfx1250) — compile-verified
//
#include <hip/hip_runtime.h>
#include <math.h>

typedef __attribute__((ext_vector_type(2))) float v2f;
typedef __attribute__((ext_vector_type(8))) float v8f;

#define BDIM 4
#define NPTS 1024
#define NROW (BDIM * NPTS) /* 4096 */
#define PADK 12
#define TOPK 16

/* workspace layout, in floats */
#define WS_STATS 0                      /* mean[9], cnt at [9] */
#define WS_C 16                         /* 9x9 pinv(cov), row-major */
#define WS_A 112                        /* centered targets, padded to 12 */
#define WS_QA (WS_A + NROW * PADK)
#define WS_W (WS_QA + NROW)             /* C*p rows, padded */
#define WS_QP (WS_W + NROW * PADK)
#define WS_U (WS_QP + NROW)             /* p+mean rows, padded */
#define WS_QU (WS_U + NROW * PADK)
#define WS_RSI (WS_QU + NROW)           /* intra per-row top-16 sums */
#define WS_RSO (WS_RSI + NROW)          /* outer per-row top-16 sums */

// ---------------------------------------------------------------------------
// K1: masked mean + covariance (deterministic tree/serial reductions) and
//     pinv via single-thread cyclic Jacobi eigendecomposition of the 9x9.
// ---------------------------------------------------------------------------
__global__ __launch_bounds__(256) void stats_kernel(const float* __restrict__ targets,
                                                    const int* __restrict__ mask,
                                                    float* __restrict__ ws) {
  __shared__ float part[256 * 45];
  __shared__ float tot[45];
  __shared__ float smean[9];
  __shared__ float scnt;
  const int tid = threadIdx.x;

  // ---- phase 1: masked sums of targets (9) + count --------------------------
  float acc[10];
#pragma unroll
  for (int j = 0; j < 10; ++j) acc[j] = 0.f;
  for (int r = tid; r < NROW; r += 256) {
    const float mk = (mask[r] != 0) ? 1.f : 0.f;
    const float* t = targets + r * 9;
#pragma unroll
    for (int j = 0; j < 9; ++j) acc[j] += mk * t[j];
    acc[9] += mk;
  }
#pragma unroll
  for (int j = 0; j < 10; ++j) part[tid * 10 + j] = acc[j];
  __syncthreads();
  if (tid < 10) {  // fixed-order serial sum -> deterministic
    float s = 0.f;
    for (int t = 0; t < 256; ++t) s += part[t * 10 + tid];
    tot[tid] = s;
  }
  __syncthreads();
  if (tid == 0) scnt = tot[9];
  if (tid < 9) smean[tid] = tot[tid] / tot[9];
  __syncthreads();

  // ---- phase 2: covariance (upper triangle, 45 entries) ---------------------
  float cov[45];
#pragma unroll
  for (int c = 0; c < 45; ++c) cov[c] = 0.f;
  float mloc[9];
#pragma unroll
  for (int j = 0; j < 9; ++j) mloc[j] = smean[j];
  for (int r = tid; r < NROW; r += 256) {
    const float mk = (mask[r] != 0) ? 1.f : 0.f;
    const float* t = targets + r * 9;
    float x[9];
#pragma unroll
    for (int j = 0; j < 9; ++j) x[j] = (t[j] - mloc[j]) * mk;
    int c = 0;
#pragma unroll
    for (int i = 0; i < 9; ++i)
#pragma unroll
      for (int j = i; j < 9; ++j, ++c) cov[c] += x[i] * x[j];
  }
  __syncthreads();
#pragma unroll
  for (int c = 0; c < 45; ++c) part[tid * 45 + c] = cov[c];
  __syncthreads();
  if (tid < 45) {
    float s = 0.f;
    for (int t = 0; t < 256; ++t) s += part[t * 45 + tid];
    tot[tid] = s;
  }
  __syncthreads();

  // ---- phase 3 (thread 0): Jacobi eigendecomposition + pseudoinverse --------
  if (tid == 0) {
    const float cnt = scnt;
    float A[9][9], V[9][9];
    int c = 0;
    for (int i = 0; i < 9; ++i)
      for (int j = i; j < 9; ++j, ++c) {
        const float v = tot[c] / (cnt - 1.f);
        A[i][j] = v;
        A[j][i] = v;
      }
    for (int i = 0; i < 9; ++i)
      for (int j = 0; j < 9; ++j) V[i][j] = (i == j) ? 1.f : 0.f;
    for (int sweep = 0; sweep < 16; ++sweep) {
      for (int p = 0; p < 8; ++p)
        for (int q = p + 1; q < 9; ++q) {
          const float apq = A[p][q];
          if (fabsf(apq) < 1e-30f) continue;
          const float theta = (A[q][q] - A[p][p]) / (2.f * apq);
          const float tt =
              ((theta >= 0.f) ? 1.f : -1.f) / (fabsf(theta) + sqrtf(theta * theta + 1.f));
          const float cth = 1.f / sqrtf(tt * tt + 1.f);
          const float sth = tt * cth;
          for (int k = 0; k < 9; ++k) {
            const float akp = A[k][p], akq = A[k][q];
            A[k][p] = cth * akp - sth * akq;
            A[k][q] = sth * akp + cth * akq;
          }
          for (int k = 0; k < 9; ++k) {
            const float apk = A[p][k], aqk = A[q][k];
            A[p][k] = cth * apk - sth * aqk;
            A[q][k] = sth * apk + cth * aqk;
          }
          for (int k = 0; k < 9; ++k) {
            const float vkp = V[k][p], vkq = V[k][q];
            V[k][p] = cth * vkp - sth * vkq;
            V[k][q] = sth * vkp + cth * vkq;
          }
        }
    }
    float dmax = 0.f;
    for (int i = 0; i < 9; ++i) dmax = fmaxf(dmax, fabsf(A[i][i]));
    const float tol = dmax * 9.f * 1.1920929e-07f;
    float invd[9];
    for (int i = 0; i < 9; ++i) invd[i] = (fabsf(A[i][i]) > tol) ? (1.f / A[i][i]) : 0.f;
    for (int i = 0; i < 9; ++i)
      for (int j = 0; j < 9; ++j) {
        float s = 0.f;
        for (int k = 0; k < 9; ++k) s += V[i][k] * invd[k] * V[j][k];
        ws[WS_C + i * 9 + j] = s;
      }
    for (int j = 0; j < 9; ++j) ws[WS_STATS + j] = smean[j];
    ws[WS_STATS + 9] = cnt;
  }
}

// ---------------------------------------------------------------------------
// K2: per-row precompute. a=t-mean, w=C*p, u=p+mean; q_a=a'Ca, q_p=p'Cp, q_u=u'Cu.
//     Rows stored padded to K=12 (zeros) for WMMA f32 16x16x4 (K multiple of 4).
// ---------------------------------------------------------------------------
__global__ __launch_bounds__(256) void precompute_kernel(const float* __restrict__ outputs,
                                                         const float* __restrict__ targets,
                                                         float* __restrict__ ws) {
  __shared__ float sC[81];
  __shared__ float sM[9];
  const int tid = threadIdx.x;
  if (tid < 81) sC[tid] = ws[WS_C + tid];
  if (tid < 9) sM[tid] = ws[WS_STATS + tid];
  __syncthreads();
  const int r = blockIdx.x * 256 + tid;
  if (r >= NROW) return;
  float p[9], t[9], a[9], u[9];
#pragma unroll
  for (int j = 0; j < 9; ++j) {
    p[j] = outputs[r * 9 + j];
    t[j] = targets[r * 9 + j];
  }
#pragma unroll
  for (int j = 0; j < 9; ++j) {
    a[j] = t[j] - sM[j];
    u[j] = p[j] + sM[j];
  }
  float w[9], Ca[9], Cu[9];
#pragma unroll
  for (int i = 0; i < 9; ++i) {
    float sw = 0.f, sa = 0.f, su = 0.f;
#pragma unroll
    for (int k = 0; k < 9; ++k) {
      const float cik = sC[i * 9 + k];
      sw += cik * p[k];
      sa += cik * a[k];
      su += cik * u[k];
    }
    w[i] = sw;
    Ca[i] = sa;
    Cu[i] = su;
  }
  float qp = 0.f, qa = 0.f, qu = 0.f;
#pragma unroll
  for (int i = 0; i < 9; ++i) {
    qp += p[i] * w[i];
    qa += a[i] * Ca[i];
    qu += u[i] * Cu[i];
  }
  float* Aout = ws + WS_A + (size_t)r * PADK;
  float* Wout = ws + WS_W + (size_t)r * PADK;
  float* Uout = ws + WS_U + (size_t)r * PADK;
#pragma unroll
  for (int j = 0; j < 9; ++j) {
    Aout[j] = a[j];
    Wout[j] = w[j];
    Uout[j] = u[j];
  }
#pragma unroll
  for (int j = 9; j < PADK; ++j) {
    Aout[j] = 0.f;
    Wout[j] = 0.f;
    Uout[j] = 0.f;
  }
  ws[WS_QA + r] = qa;
  ws[WS_QP + r] = qp;
  ws[WS_QU + r] = qu;
}

// ---------------------------------------------------------------------------
// K3: pairwise distance slab (16 rows x 1024 cols) via V_WMMA_F32_16X16X4_F32
//     (3 chained WMMAs, K padded 9->12), d = q_row + q_col - 2*G, written to
//     LDS, then per-row sum of the 16 smallest via wave32 shuffle argmin.
//     d(row,col) = q_row[row] + q_col[col] - 2 * <rowvec[row], colvec[col]>.
// ---------------------------------------------------------------------------
__global__ __launch_bounds__(128) void pairwise_topk_kernel(const float* __restrict__ rowvec,
                                                            const float* __restrict__ rowq,
                                                            const float* __restrict__ colvec,
                                                            const float* __restrict__ colq,
                                                            float* __restrict__ rowsum) {
  __shared__ float ds[16][1024];  // 64 KB of the 320 KB WGP LDS
  const int b = blockIdx.x >> 6;
  const int tile = blockIdx.x & 63;
  const int rowbase = b * NPTS + tile * 16;
  const int wave = threadIdx.x >> 5;
  const int lane = threadIdx.x & 31;
  const int half = lane >> 4;   // wave32: lanes 0-15 vs 16-31
  const int nloc = lane & 15;

  // A fragments (16x4 f32): lane holds row M=lane%16, K pair {2*half, 2*half+1}
  v2f afrag[3];
  {
    const float* ar = rowvec + (size_t)(rowbase + nloc) * PADK;
#pragma unroll
    for (int ka = 0; ka < 3; ++ka) {
      const int c0 = 4 * ka + 2 * half;
      afrag[ka].x = ar[c0];
      afrag[ka].y = ar[c0 + 1];
    }
  }
  float rq[8];
#pragma unroll
  for (int r = 0; r < 8; ++r) rq[r] = rowq[rowbase + r + 8 * half];

  // each wave walks 16 of the 64 column tiles; EXEC all-ones throughout WMMA
  for (int ct = wave; ct < 64; ct += 4) {
    const int colbase = b * NPTS + ct * 16;
    const float* wr = colvec + (size_t)(colbase + nloc) * PADK;
    v2f bfrag[3];  // B (4x16) = colvec^T: lane holds col N=lane%16, same K pair
#pragma unroll
    for (int kb = 0; kb < 3; ++kb) {
      const int c0 = 4 * kb + 2 * half;
      bfrag[kb].x = wr[c0];
      bfrag[kb].y = wr[c0 + 1];
    }
    const float cq = colq[colbase + nloc];
    v8f c = {0.f, 0.f, 0.f, 0.f, 0.f, 0.f, 0.f, 0.f};
    c = __builtin_amdgcn_wmma_f32_16x16x4_f32(false, afrag[0], false, bfrag[0], (short)0, c,
                                              false, false);
    c = __builtin_amdgcn_wmma_f32_16x16x4_f32(false, afrag[1], false, bfrag[1], (short)0, c,
                                              false, false);
    c = __builtin_amdgcn_wmma_f32_16x16x4_f32(false, afrag[2], false, bfrag[2], (short)0, c,
                                              false, false);
    // C/D layout: lane half selects M base (0 or 8), VGPR index r is M offset, N=lane%16
#pragma unroll
    for (int r = 0; r < 8; ++r)
      ds[r + 8 * half][ct * 16 + nloc] = rq[r] + cq - 2.f * c[r];
  }
  __syncthreads();

  // selection: one wave per row (4 rows per wave), 32 values per lane in regs,
  // 16 x (local scan + 5-step shuffle argmin with (value,lane) tie-break).
  for (int rr = 0; rr < 4; ++rr) {
    const int row = wave * 4 + rr;
    float v[32];
#pragma unroll
    for (int k = 0; k < 32; ++k) v[k] = ds[row][k * 32 + lane];
    float sum = 0.f;
    for (int it = 0; it < TOPK; ++it) {
      float lm = v[0];
      int li = 0;
#pragma unroll
      for (int k = 1; k < 32; ++k)
        if (v[k] < lm) {
          lm = v[k];
          li = k;
        }
      float mval = lm;
      int src = lane;
#pragma unroll
      for (int off = 16; off > 0; off >>= 1) {
        const float om = __shfl_xor(mval, off, 32);
        const int os = __shfl_xor(src, off, 32);
        if (om < mval || (om == mval && os < src)) {
          mval = om;
          src = os;
        }
      }
      sum += mval;
      if (lane == src) {
#pragma unroll
        for (int k = 0; k < 32; ++k)
          if (k == li) v[k] = 3.0e38f;
      }
    }
    if (lane == 0) rowsum[rowbase + row] = sum;
  }
}

// ---------------------------------------------------------------------------
// K4: final deterministic reductions -> 3 output scalars.
// ---------------------------------------------------------------------------
__global__ __launch_bounds__(256) void final_kernel(const int* __restrict__ mask,
                                                    const float* __restrict__ ws,
                                                    float* __restrict__ out) {
  __shared__ float pi[256], po[256];
  const int tid = threadIdx.x;
  float si = 0.f, so = 0.f;
  for (int r = tid; r < NROW; r += 256) {
    si += ws[WS_RSI + r] * ((mask[r] != 0) ? 1.f : 0.f);
    so += ws[WS_RSO + r];
  }
  pi[tid] = si;
  po[tid] = so;
  __syncthreads();
  if (tid == 0) {
    float ti = 0.f, to = 0.f;
    for (int t = 0; t < 256; ++t) {
      ti += pi[t];
      to += po[t];
    }
    const float cnt = ws[WS_STATS + 9];
    const float intra = ti / cnt;
    const float outer = to / ((float)NROW * (float)TOPK);
    out[0] = intra;
    out[1] = intra;
    out[2] = outer;
  }
}

extern "C" void kernel_launch(void* const* d_in, const int* in_sizes, int n_in,
                              void* d_out, int out_size, void* d_ws, size_t ws_size,
                              hipStream_t stream) {
  const float* outputs = (const float*)d_in[0];  // (4,1024,9) f32
  const float* targets = (const float*)d_in[1];  // (4,1024,9) f32
  const int* mask = (const int*)d_in[2];         // (4,1024) int 0/1
  float* ws = (float*)d_ws;                      // needs ~672 KB
  float* out = (float*)d_out;                    // 3 floats

  stats_kernel<<<1, 256, 0, stream>>>(targets, mask, ws);
  precompute_kernel<<<NROW / 256, 256, 0, stream>>>(outputs, targets, ws);
  // intra: rows = (t - mean), cols = C*p ; d over (b, m, n)
  pairwise_topk_kernel<<<BDIM * 64, 128, 0, stream>>>(ws + WS_A, ws + WS_QA, ws + WS_W,
                                                      ws + WS_QP, ws + WS_RSI);
  // outer: rows = (p + mean), cols = C*p ; d over (b, i, j)
  pairwise_topk_kernel<<<BDIM * 64, 128, 0, stream>>>(ws + WS_U, ws + WS_QU, ws + WS_W,
                                                      ws + WS_QP, ws + WS_RSO);
  final_kernel<<<1, 256, 0, stream>>>(mask, ws, out);
}